// DifferentiableGrowingNeuralGas_61469571940955
// MI455X (gfx1250) — compile-verified
//
#include <hip/hip_runtime.h>
#include <hip/hip_bf16.h>

// Sizes fixed by the reference
#define BN 4096
#define MN 256
#define DN 256

#define TAU_INV 5.0f        // 1/0.2
#define LAM_INV 0.125f      // 1/8.0
#define TOPO 0.5f
#define LEN_C 0.01f
#define SP_C 0.001f

typedef float v2f __attribute__((ext_vector_type(2)));
typedef float v8f __attribute__((ext_vector_type(8)));

// ---------------------------------------------------------------------------
// Row squared-norms: blocks [0,BN) -> data rows, [BN,BN+MN) -> weight rows
// ---------------------------------------------------------------------------
__global__ __launch_bounds__(256) void norms_kernel(
    const float* __restrict__ data, const float* __restrict__ wts,
    float* __restrict__ dnorm, float* __restrict__ wnorm)
{
    __shared__ float s[256];
    int b = blockIdx.x;
    const float* src;
    float* dst;
    int row;
    if (b < BN) { src = data; dst = dnorm; row = b; }
    else        { src = wts;  dst = wnorm; row = b - BN; }
    float x = src[row * DN + threadIdx.x];
    s[threadIdx.x] = x * x;
    __syncthreads();
    for (int off = 128; off > 0; off >>= 1) {
        if (threadIdx.x < (unsigned)off) s[threadIdx.x] += s[threadIdx.x + off];
        __syncthreads();
    }
    if (threadIdx.x == 0) dst[row] = s[0];
}

// ---------------------------------------------------------------------------
// out[i][j] = ||A_i - B_j||^2 (clamped >= 0), optionally sqrt'ed.
// dot(A_i, B_j) computed with V_WMMA_F32_16X16X4_F32, one wave per 16x16 tile.
//
// f32 WMMA fragment layout (ISA 7.12.2):
//   A 16x4:  lane L holds row M=L%16; VGPR0 = K = (L<16 ? 0 : 2), VGPR1 = K+1
//   B 4x16:  lane L holds col N=L%16; VGPR0 = K = (L<16 ? 0 : 2), VGPR1 = K+1
//            where B[k][n] = Bm[n][k]  (Bm row-major, we need Bm^T)
//   C/D 16x16: VGPR v, lane L -> M = v + (L<16 ? 0 : 8), N = L%16
// ---------------------------------------------------------------------------
__global__ __launch_bounds__(32) void sqdist_wmma_kernel(
    const float* __restrict__ A, const float* __restrict__ Bm,
    const float* __restrict__ normA, const float* __restrict__ normB,
    float* __restrict__ out, int K, int nTilesN, int doSqrt)
{
    const int tile = blockIdx.x;
    const int tm = tile / nTilesN;
    const int tn = tile % nTilesN;
    const int lane = threadIdx.x;      // 0..31, full wave, EXEC all ones
    const int half = lane >> 4;        // 0 or 1
    const int l    = lane & 15;
    const int koff = half * 2;

    const float* aRow = A  + (size_t)(tm * 16 + l) * K + koff;  // A row M=l
    const float* bRow = Bm + (size_t)(tn * 16 + l) * K + koff;  // B col N=l

    v8f c = {};
#pragma unroll 8
    for (int k0 = 0; k0 < K; k0 += 4) {
        v2f a = *(const v2f*)(aRow + k0);  // 8B-aligned (k0%4==0, koff in {0,2})
        v2f b = *(const v2f*)(bRow + k0);
        c = __builtin_amdgcn_wmma_f32_16x16x4_f32(
            /*neg_a=*/false, a, /*neg_b=*/false, b,
            /*c_mod=*/(short)0, c, /*reuse_a=*/false, /*reuse_b=*/false);
    }

    const int n  = tn * 16 + l;
    const int N  = nTilesN * 16;
    const float nb = normB[n];
#pragma unroll
    for (int v = 0; v < 8; ++v) {
        int m = tm * 16 + v + half * 8;
        float d2 = fmaxf(normA[m] - 2.0f * c[v] + nb, 0.0f);
        out[(size_t)m * N + n] = doSqrt ? sqrtf(d2) : d2;
    }
}

// ---------------------------------------------------------------------------
// Edge terms: ep = sigmoid(0.5*(L[i][j]+L[j][i])), zero diagonal.
// degree[i] = sum_j ep / (M-1); per-row partials of sum(ep) and sum(ep*pd).
// ---------------------------------------------------------------------------
__global__ __launch_bounds__(256) void edge_kernel(
    const float* __restrict__ el, const float* __restrict__ proto,
    float* __restrict__ degree, float* __restrict__ epPart,
    float* __restrict__ wlPart)
{
    __shared__ float sE[256];
    __shared__ float sW[256];
    const int i = blockIdx.x;
    const int j = threadIdx.x;

    float lij = 0.5f * (el[i * MN + j] + el[j * MN + i]);
    float ep  = 1.0f / (1.0f + __expf(-lij));
    if (j == i) ep = 0.0f;
    float pd = proto[i * MN + j];  // squared L2 between prototypes

    sE[j] = ep;
    sW[j] = ep * pd;
    __syncthreads();
    for (int off = 128; off > 0; off >>= 1) {
        if (j < off) { sE[j] += sE[j + off]; sW[j] += sW[j + off]; }
        __syncthreads();
    }
    if (j == 0) {
        degree[i] = sE[0] / (float)(MN - 1);
        epPart[i] = sE[0];
        wlPart[i] = sW[0];
    }
}

// ---------------------------------------------------------------------------
// Soft rank + data term. One block per batch row; distances row in LDS.
// sum_{j != i} sigmoid((d_i - d_j)/tau) = (sum over all j) - 0.5
// neighborhood = exp(-sum/lam) * (1 + topo*degree[i]); partial = sum_i nb*d_i
// ---------------------------------------------------------------------------
__global__ __launch_bounds__(256) void softrank_kernel(
    const float* __restrict__ dist, const float* __restrict__ degree,
    float* __restrict__ dataPart)
{
    __shared__ float sd[256];
    __shared__ float sr[256];
    const int b = blockIdx.x;
    const int i = threadIdx.x;
    const float di = dist[(size_t)b * MN + i];
    sd[i] = di;
    __syncthreads();

    float acc = 0.0f;
#pragma unroll 8
    for (int j = 0; j < MN; ++j) {
        float x = (di - sd[j]) * TAU_INV;
        // sigmoid via v_exp_f32 + v_rcp_f32
        acc += __builtin_amdgcn_rcpf(1.0f + __expf(-x));
    }
    acc -= 0.5f;  // remove the j==i term (sigmoid(0) = 0.5)

    float nb = __expf(-acc * LAM_INV);
    nb *= (1.0f + TOPO * degree[i]);

    sr[i] = nb * di;
    __syncthreads();
    for (int off = 128; off > 0; off >>= 1) {
        if (i < off) sr[i] += sr[i + off];
        __syncthreads();
    }
    if (i == 0) dataPart[b] = sr[0];
}

// ---------------------------------------------------------------------------
// Deterministic final reduction -> scalar loss
// ---------------------------------------------------------------------------
__global__ __launch_bounds__(256) void finalize_kernel(
    const float* __restrict__ dataPart, const float* __restrict__ epPart,
    const float* __restrict__ wlPart, float* __restrict__ out)
{
    __shared__ float s[256];
    const int t = threadIdx.x;

    float a = 0.0f;
    for (int k = t; k < BN; k += 256) a += dataPart[k];  // fixed order
    s[t] = a;
    __syncthreads();
    for (int off = 128; off > 0; off >>= 1) {
        if (t < off) s[t] += s[t + off];
        __syncthreads();
    }
    float dataSum = s[0];
    __syncthreads();

    s[t] = epPart[t];
    __syncthreads();
    for (int off = 128; off > 0; off >>= 1) {
        if (t < off) s[t] += s[t + off];
        __syncthreads();
    }
    float epSum = s[0];
    __syncthreads();

    s[t] = wlPart[t];
    __syncthreads();
    for (int off = 128; off > 0; off >>= 1) {
        if (t < off) s[t] += s[t + off];
        __syncthreads();
    }
    float wlSum = s[0];

    if (t == 0) {
        float data_term    = dataSum / ((float)BN * (float)MN);
        float weighted_len = wlSum / (epSum + 1e-8f);
        float sparsity     = epSum / ((float)MN * (float)MN);
        out[0] = data_term + LEN_C * weighted_len + SP_C * sparsity;
    }
}

// ---------------------------------------------------------------------------
extern "C" void kernel_launch(void* const* d_in, const int* in_sizes, int n_in,
                              void* d_out, int out_size, void* d_ws, size_t ws_size,
                              hipStream_t stream) {
    const float* data = (const float*)d_in[0];  // [BN, DN]
    const float* wts  = (const float*)d_in[1];  // [MN, DN]
    const float* el   = (const float*)d_in[2];  // [MN, MN]
    float* out = (float*)d_out;

    // Workspace layout (floats), ~4.5 MB total
    float* ws       = (float*)d_ws;
    float* dist     = ws;                         // BN*MN
    float* proto    = dist  + (size_t)BN * MN;    // MN*MN (squared dists)
    float* dnorm    = proto + (size_t)MN * MN;    // BN
    float* wnorm    = dnorm + BN;                 // MN
    float* degree   = wnorm + MN;                 // MN
    float* dataPart = degree + MN;                // BN
    float* epPart   = dataPart + BN;              // MN
    float* wlPart   = epPart + MN;                // MN

    // 1) row norms for data and weights
    norms_kernel<<<BN + MN, 256, 0, stream>>>(data, wts, dnorm, wnorm);

    // 2) distances[B,M] = sqrt(||data_b - w_m||^2) via f32 WMMA
    sqdist_wmma_kernel<<<(BN / 16) * (MN / 16), 32, 0, stream>>>(
        data, wts, dnorm, wnorm, dist, DN, MN / 16, /*doSqrt=*/1);

    // 3) proto_dist[M,M] = ||w_i - w_j||^2 via f32 WMMA (no sqrt)
    sqdist_wmma_kernel<<<(MN / 16) * (MN / 16), 32, 0, stream>>>(
        wts, wts, wnorm, wnorm, proto, DN, MN / 16, /*doSqrt=*/0);

    // 4) edge probabilities, degree, edge-term partials
    edge_kernel<<<MN, 256, 0, stream>>>(el, proto, degree, epPart, wlPart);

    // 5) soft rank + neighborhood-weighted data term partials
    softrank_kernel<<<BN, 256, 0, stream>>>(dist, degree, dataPart);

    // 6) deterministic reduction to the scalar loss
    finalize_kernel<<<1, 256, 0, stream>>>(dataPart, epPart, wlPart, out);
}